// DMPNNEncoder_64819646431794
// MI455X (gfx1250) — compile-verified
//
#include <hip/hip_runtime.h>

typedef __attribute__((ext_vector_type(2))) float v2f;
typedef __attribute__((ext_vector_type(8))) float v8f;

#define NN 16384
#define DD 128

// ---------------------------------------------------------------------------
// Kernel 0: zero col_w (NN) + m (128) + mw (128)  (contiguous region in ws)
// ---------------------------------------------------------------------------
__global__ void k_zero(float* __restrict__ p, int n) {
    int t = blockIdx.x * blockDim.x + threadIdx.x;
    if (t < n) p[t] = 0.0f;
}

// ---------------------------------------------------------------------------
// Kernel 1: col_w[n] = sum_a bond_n[a, n]  (1 GiB stream -> BW bound)
// grid = (NN/1024, 128), block = 256; each thread: 4 columns, 128 rows.
// ---------------------------------------------------------------------------
__global__ __launch_bounds__(256)
void k_colsum(const float* __restrict__ bond_n, float* __restrict__ col_w) {
    const int col  = (blockIdx.x * 256 + threadIdx.x) * 4;
    const int row0 = blockIdx.y * 128;
    const float4* p = (const float4*)(bond_n + (size_t)row0 * NN + col);
    float4 acc = make_float4(0.f, 0.f, 0.f, 0.f);
#pragma unroll 8
    for (int r = 0; r < 128; ++r) {
        float4 v = p[(size_t)r * (NN / 4)];
        acc.x += v.x; acc.y += v.y; acc.z += v.z; acc.w += v.w;
    }
    atomicAdd(&col_w[col + 0], acc.x);
    atomicAdd(&col_w[col + 1], acc.y);
    atomicAdd(&col_w[col + 2], acc.z);
    atomicAdd(&col_w[col + 3], acc.w);
}

// ---------------------------------------------------------------------------
// Kernel 2: cat[n, 0:128]   = atom_emb[x[n,0]] + atom_emb[x[n,1]]
//           cat[n, 128:256] = bond_emb[ea[n,0]] + bond_emb[ea[n,1]]
// ---------------------------------------------------------------------------
__global__ __launch_bounds__(256)
void k_embed(const int* __restrict__ x, const int* __restrict__ ea,
             const float* __restrict__ atom_emb, const float* __restrict__ bond_emb,
             float* __restrict__ cat) {
    int t = blockIdx.x * 256 + threadIdx.x;   // t in [0, NN*128)
    int n = t >> 7;
    int d = t & 127;
    float av = atom_emb[x[2 * n] * DD + d] + atom_emb[x[2 * n + 1] * DD + d];
    float bv = bond_emb[ea[2 * n] * DD + d] + bond_emb[ea[2 * n + 1] * DD + d];
    cat[(size_t)n * 256 + d]       = av;
    cat[(size_t)n * 256 + 128 + d] = bv;
}

// ---------------------------------------------------------------------------
// Kernel 3: h_n = relu(cat @ Wi_w[2].T + Wi_b[2])   [NN,128] fp32 WMMA GEMM
// One wave computes one 16x16 output tile; K=256 -> 64 x V_WMMA_F32_16X16X4_F32.
// Lane layout (ISA 7.12.2, wave32):
//   A 16x4 : a[v] = A[M = lane%16][K = (lane/16)*2 + v]
//   B 4x16 : b[v] = B[K = (lane/16)*2 + v][N = lane%16]
//   C 16x16: c[r] = C[M = (lane/16)*8 + r][N = lane%16]
// ---------------------------------------------------------------------------
__global__ __launch_bounds__(128)
void k_gemm_hn(const float* __restrict__ cat, const float* __restrict__ Wi_w,
               const float* __restrict__ Wi_b, float* __restrict__ h_n) {
    const int wave  = threadIdx.x >> 5;
    const int lane  = threadIdx.x & 31;
    const int tile  = blockIdx.x * 4 + wave;   // 8192 tiles total
    const int tileM = tile >> 3;               // 1024 row tiles
    const int tileN = tile & 7;                // 8 col tiles
    const int half  = lane >> 4;
    const int l16   = lane & 15;
    const int row   = tileM * 16 + l16;        // A row (M)
    const int col   = tileN * 16 + l16;        // B/C col (N)

    // layer 2 of Wi_w: [128][256], row j holds weights for output feature j
    const float* W2    = Wi_w + 2 * DD * (2 * DD);
    const float* aBase = cat + (size_t)row * 256 + half * 2;  // K stride 1
    const float* bBase = W2  + (size_t)col * 256 + half * 2;  // B[k][col] = W2[col][k]

    v8f c = {};
#pragma unroll 8
    for (int k0 = 0; k0 < 256; k0 += 4) {
        v2f a = *(const v2f*)(aBase + k0);     // K = k0 + half*2 + {0,1}
        v2f b = *(const v2f*)(bBase + k0);
        c = __builtin_amdgcn_wmma_f32_16x16x4_f32(
                /*neg_a=*/false, a, /*neg_b=*/false, b,
                /*c_mod=*/(short)0, c, /*reuse_a=*/false, /*reuse_b=*/false);
    }

    const float bias = Wi_b[2 * DD + col];
#pragma unroll
    for (int r = 0; r < 8; ++r) {
        int M = tileM * 16 + half * 8 + r;
        float v = c[r] + bias;
        h_n[(size_t)M * DD + col] = v > 0.f ? v : 0.f;
    }
}

// ---------------------------------------------------------------------------
// Kernel 4: m[j] = sum_n col_w[n] * h_n[n, j]
// grid = 128 row-chunks, block = 128 (one thread per feature j)
// ---------------------------------------------------------------------------
__global__ __launch_bounds__(128)
void k_mreduce(const float* __restrict__ col_w, const float* __restrict__ h_n,
               float* __restrict__ m) {
    const int j  = threadIdx.x;
    const int n0 = blockIdx.x * 128;
    float acc = 0.f;
#pragma unroll 4
    for (int r = 0; r < 128; ++r) {
        int n = n0 + r;
        acc += col_w[n] * h_n[(size_t)n * DD + j];
    }
    atomicAdd(&m[j], acc);
}

// ---------------------------------------------------------------------------
// Kernel 5: mw[j] = sum_k m[k] * Wm_w[2][j][k] + Wm_b[2][j]   (tiny, 1 block)
// ---------------------------------------------------------------------------
__global__ __launch_bounds__(128)
void k_mw(const float* __restrict__ m, const float* __restrict__ Wm_w,
          const float* __restrict__ Wm_b, float* __restrict__ mw) {
    const int j = threadIdx.x;
    const float* W2 = Wm_w + 2 * DD * DD + (size_t)j * DD;
    float s = Wm_b[2 * DD + j];
#pragma unroll 8
    for (int k = 0; k < DD; ++k) s += m[k] * W2[k];
    mw[j] = s;
}

// ---------------------------------------------------------------------------
// Kernel 6: h[n,j] = relu(h_n[n,j] + mw[j])
// ---------------------------------------------------------------------------
__global__ __launch_bounds__(256)
void k_final(const float* __restrict__ h_n, const float* __restrict__ mw,
             float* __restrict__ h) {
    int t = blockIdx.x * 256 + threadIdx.x;   // t in [0, NN*128)
    int j = t & 127;
    float v = h_n[t] + mw[j];
    h[t] = v > 0.f ? v : 0.f;
}

// ---------------------------------------------------------------------------
extern "C" void kernel_launch(void* const* d_in, const int* in_sizes, int n_in,
                              void* d_out, int out_size, void* d_ws, size_t ws_size,
                              hipStream_t stream) {
    const int*   x        = (const int*)  d_in[0];   // [N,2]
    const int*   ea       = (const int*)  d_in[1];   // [N,2]
    const float* bond_n   = (const float*)d_in[2];   // [A,N]
    const float* atom_emb = (const float*)d_in[3];   // [120,128]
    const float* bond_emb = (const float*)d_in[4];   // [6,128]
    const float* Wi_w     = (const float*)d_in[5];   // [3,128,256]
    const float* Wi_b     = (const float*)d_in[6];   // [3,128]
    const float* Wm_w     = (const float*)d_in[7];   // [3,128,128]
    const float* Wm_b     = (const float*)d_in[8];   // [3,128]
    float*       h_out    = (float*)d_out;           // [N,128]

    // workspace layout (floats): cat | h_n | col_w | m | mw  (last three contiguous)
    float* ws    = (float*)d_ws;
    float* cat   = ws;                                 // NN*256
    float* h_n   = cat + (size_t)NN * 256;             // NN*128
    float* col_w = h_n + (size_t)NN * DD;              // NN
    float* m     = col_w + NN;                         // 128
    float* mw    = m + DD;                             // 128

    // 0: zero col_w + m + mw
    {
        int n = NN + 2 * DD;
        k_zero<<<(n + 255) / 256, 256, 0, stream>>>(col_w, n);
    }
    // 1: column sums of bond_n
    {
        dim3 grid(NN / (256 * 4), NN / 128);           // (16, 128)
        k_colsum<<<grid, 256, 0, stream>>>(bond_n, col_w);
    }
    // 2: build cat
    k_embed<<<(NN * DD) / 256, 256, 0, stream>>>(x, ea, atom_emb, bond_emb, cat);
    // 3: WMMA GEMM -> h_n (fused bias + relu)
    k_gemm_hn<<<(NN / 16) * (DD / 16) / 4, 128, 0, stream>>>(cat, Wi_w, Wi_b, h_n);
    // 4: m = col_w @ h_n
    k_mreduce<<<NN / 128, 128, 0, stream>>>(col_w, h_n, m);
    // 5: mw = m @ Wm_w[2].T + Wm_b[2]
    k_mw<<<1, 128, 0, stream>>>(m, Wm_w, Wm_b, mw);
    // 6: h = relu(h_n + mw)
    k_final<<<(NN * DD) / 256, 256, 0, stream>>>(h_n, mw, h_out);
}